// NeuroQuantLoss_84095459656193
// MI455X (gfx1250) — compile-verified
//
#include <hip/hip_runtime.h>

typedef __attribute__((ext_vector_type(2))) float v2f;
typedef __attribute__((ext_vector_type(8))) float v8f;

#define BATCH 4
#define DDIM 160
#define HDIM 192
#define WDIM 192
#define NTOT (BATCH * DDIM * HDIM * WDIM)   // 23,592,960

__device__ __forceinline__ float gauss(int j) {
    // gaussian_1d(window=7, sigma=1.5), normalized
    constexpr float g[7] = {0.03663285f, 0.11128076f, 0.21674532f, 0.27068214f,
                            0.21674532f, 0.11128076f, 0.03663285f};
    return g[j];
}

// CDNA5 async global->LDS copy (ASYNCcnt tracked, no VGPR round-trip).
__device__ __forceinline__ void async_load_f32(const float* gptr, float* lptr) {
    const unsigned lds = (unsigned)(uintptr_t)lptr;            // LDS byte offset
    const unsigned long long ga = (unsigned long long)(uintptr_t)gptr;
    asm volatile("global_load_async_to_lds_b32 %0, %1, off"
                 :: "v"(lds), "v"(ga) : "memory");
}

__device__ __forceinline__ void wait_asynccnt0() {
    asm volatile("s_wait_asynccnt 0x0" ::: "memory");
}

__global__ void nq_init_kernel(float* acc) {
    acc[0] = 0.f;  // sum w*|r-t|
    acc[1] = 0.f;  // sum w
    acc[2] = 0.f;  // sum ssim_map
}

__global__ __launch_bounds__(256) void nq_main_kernel(
        const float* __restrict__ recon,
        const float* __restrict__ target,
        float* __restrict__ acc) {
    // Block: 16x16 (H,W) core tile, marching D with:
    //  - double-buffered raw halo slabs filled by async global->LDS DMA
    //  - W-blur on VALU, H-blur as banded GEMM on the WMMA pipe
    //  - 7-slice D ring buffer in LDS for the final D-blur
    __shared__ float bufX[2][22 * 22];
    __shared__ float bufY[2][22 * 22];
    __shared__ float t1[5][22][16];       // after W-blur (H still halo)
    __shared__ float ring[7][5][256];     // after W+H blur, 7-deep D ring
    __shared__ float red[3][8];

    const int t    = threadIdx.x;
    const int lane = t & 31;
    const int wave = t >> 5;
    const int b     = blockIdx.z;
    const int tileH = blockIdx.y * 16;
    const int tileW = blockIdx.x * 16;

    constexpr float C1 = 0.0004f;  // (0.01*2)^2
    constexpr float C2 = 0.0036f;  // (0.03*2)^2

    // ---- A-matrix (Gaussian band G[16x24]) lane fragments for the WMMA
    //      H-blur: out(16x16) = G(16x22, zero-padded to K=24) x t1(22x16).
    const int m      = lane & 15;
    const int hiHalf = lane >> 4;
    float aw[6][2];
#pragma unroll
    for (int c = 0; c < 6; ++c) {
        const int kbase = c * 4 + (hiHalf ? 2 : 0);
#pragma unroll
        for (int s = 0; s < 2; ++s) {
            const int dlt = (kbase + s) - m;   // band offset, valid 0..6
            float v = 0.f;
#pragma unroll
            for (int j = 0; j < 7; ++j) v = (dlt == j) ? gauss(j) : v;
            aw[c][s] = v;
        }
    }

    // ---- Zero-fill out-of-bounds halo positions once (both buffers).
    //      Async loads only ever write in-bounds positions, so zeros persist,
    //      giving SAME zero padding for free on every D-step.
    for (int i = t; i < 22 * 22; i += 256) {
        const int hh = i / 22, ww = i - hh * 22;
        const int gh = tileH + hh - 3;
        const int gw = tileW + ww - 3;
        if (!(gh >= 0 && gh < HDIM && gw >= 0 && gw < WDIM)) {
            bufX[0][i] = 0.f; bufX[1][i] = 0.f;
            bufY[0][i] = 0.f; bufY[1][i] = 0.f;
        }
    }

    // ---- Prefetch slice 0 (async DMA into buffer 0).
    for (int i = t; i < 22 * 22; i += 256) {
        const int hh = i / 22, ww = i - hh * 22;
        const int gh = tileH + hh - 3;
        const int gw = tileW + ww - 3;
        if (gh >= 0 && gh < HDIM && gw >= 0 && gw < WDIM) {
            const size_t idx = (((size_t)(b * DDIM + 0) * HDIM + gh) * WDIM + gw);
            async_load_f32(recon + idx,  &bufX[0][i]);
            async_load_f32(target + idx, &bufY[0][i]);
        }
    }

    float ssim_sum = 0.f, wl1_sum = 0.f, w_sum = 0.f;

    for (int d = 0; d < DDIM + 3; ++d) {
        const int pb = d & 1;
        if (d < DDIM) wait_asynccnt0();   // my async writes for slice d landed
        __syncthreads();                  // ... and everyone else's too
        if (d < DDIM) {
            const float* sx = bufX[pb];
            const float* sy = bufY[pb];

            // -------- weighted L1 on the 16x16 core of slice d
            {
                const int hh = (t >> 4) + 3, ww = (t & 15) + 3;
                const float r  = sx[hh * 22 + ww];
                const float tg = sy[hh * 22 + ww];
                const float w5 = (tg > -0.9f) ? 5.0f : 1.0f;
                wl1_sum += w5 * fabsf(r - tg);
                w_sum   += w5;
            }

            // -------- software pipeline: async-prefetch slice d+1 into the
            //          other buffer; it streams in while we compute below.
            if (d + 1 < DDIM) {
                for (int i = t; i < 22 * 22; i += 256) {
                    const int hh = i / 22, ww = i - hh * 22;
                    const int gh = tileH + hh - 3;
                    const int gw = tileW + ww - 3;
                    if (gh >= 0 && gh < HDIM && gw >= 0 && gw < WDIM) {
                        const size_t idx =
                            (((size_t)(b * DDIM + (d + 1)) * HDIM + gh) * WDIM + gw);
                        async_load_f32(recon + idx,  &bufX[pb ^ 1][i]);
                        async_load_f32(target + idx, &bufY[pb ^ 1][i]);
                    }
                }
            }

            // -------- pass 1: blur along W; form x,y,x^2,y^2,xy per tap
            for (int i = t; i < 22 * 16; i += 256) {
                const int h = i >> 4, w = i & 15;
                const int base = h * 22 + w;
                float v0 = 0.f, v1 = 0.f, v2 = 0.f, v3 = 0.f, v4 = 0.f;
#pragma unroll
                for (int j = 0; j < 7; ++j) {
                    const float g = gauss(j);
                    const float a = sx[base + j];
                    const float bb2 = sy[base + j];
                    v0 = fmaf(g, a, v0);
                    v1 = fmaf(g, bb2, v1);
                    v2 = fmaf(g, a * a, v2);
                    v3 = fmaf(g, bb2 * bb2, v3);
                    v4 = fmaf(g, a * bb2, v4);
                }
                t1[0][h][w] = v0;
                t1[1][h][w] = v1;
                t1[2][h][w] = v2;
                t1[3][h][w] = v3;
                t1[4][h][w] = v4;
            }
        }
        __syncthreads();

        // -------- pass 2: blur along H as banded GEMM on the WMMA pipe.
        //          One wave per field: D = G(16x24) x t1(24x16), 6x K=4.
        if (d < DDIM && wave < 5) {
            const int q = wave;
            v8f dacc = {};
#pragma unroll
            for (int c = 0; c < 6; ++c) {
                const int kbase = c * 4 + (hiHalf ? 2 : 0);
                const float b0 = (kbase     < 22) ? t1[q][kbase][m]     : 0.f;
                const float b1 = (kbase + 1 < 22) ? t1[q][kbase + 1][m] : 0.f;
                v2f A = {aw[c][0], aw[c][1]};
                v2f B = {b0, b1};
                dacc = __builtin_amdgcn_wmma_f32_16x16x4_f32(
                    false, A, false, B, (short)0, dacc, false, false);
            }
            const int slot = d % 7;
#pragma unroll
            for (int j = 0; j < 8; ++j) {
                const int h = j + (hiHalf ? 8 : 0);   // C/D layout: VGPR j -> rows j, j+8
                ring[slot][q][h * 16 + m] = dacc[j];
            }
        }
        __syncthreads();

        // -------- pass 3: blur along D from ring + SSIM map
        const int dc = d - 3;
        if (dc >= 0) {
            float s0 = 0.f, s1 = 0.f, s2 = 0.f, s3 = 0.f, s4 = 0.f;
#pragma unroll
            for (int k = 0; k < 7; ++k) {
                const int dd = dc + k - 3;
                if (dd >= 0 && dd < DDIM) {        // zero padding along D
                    const int slot = dd % 7;
                    const float gw = gauss(k);
                    s0 = fmaf(gw, ring[slot][0][t], s0);
                    s1 = fmaf(gw, ring[slot][1][t], s1);
                    s2 = fmaf(gw, ring[slot][2][t], s2);
                    s3 = fmaf(gw, ring[slot][3][t], s3);
                    s4 = fmaf(gw, ring[slot][4][t], s4);
                }
            }
            const float mx = s0, my = s1;
            const float mx2 = mx * mx, my2 = my * my, mxy = mx * my;
            const float sx2 = s2 - mx2, sy2 = s3 - my2, sxy = s4 - mxy;
            const float num = (2.f * mxy + C1) * (2.f * sxy + C2);
            const float den = (mx2 + my2 + C1) * (sx2 + sy2 + C2);
            ssim_sum += num / den;
        }
    }

    // ---------- block reduction (wave32 shuffles, then 8 waves via LDS)
#pragma unroll
    for (int off = 16; off > 0; off >>= 1) {
        ssim_sum += __shfl_down(ssim_sum, off, 32);
        wl1_sum  += __shfl_down(wl1_sum,  off, 32);
        w_sum    += __shfl_down(w_sum,    off, 32);
    }
    if (lane == 0) {
        red[0][wave] = wl1_sum;
        red[1][wave] = w_sum;
        red[2][wave] = ssim_sum;
    }
    __syncthreads();
    if (t == 0) {
        float a = 0.f, bb = 0.f, cc = 0.f;
#pragma unroll
        for (int i = 0; i < 8; ++i) {
            a += red[0][i]; bb += red[1][i]; cc += red[2][i];
        }
        atomicAdd(&acc[0], a);
        atomicAdd(&acc[1], bb);
        atomicAdd(&acc[2], cc);
    }
}

__global__ void nq_final_kernel(const float* __restrict__ acc,
                                const float* __restrict__ vq,
                                float* __restrict__ out) {
    const float l1        = acc[0] / acc[1];
    const float ssim_mean = acc[2] / (float)NTOT;
    out[0] = l1 + 0.5f * (1.0f - ssim_mean) + vq[0];
}

extern "C" void kernel_launch(void* const* d_in, const int* in_sizes, int n_in,
                              void* d_out, int out_size, void* d_ws, size_t ws_size,
                              hipStream_t stream) {
    const float* recon  = (const float*)d_in[0];
    const float* target = (const float*)d_in[1];
    const float* vq     = (const float*)d_in[2];
    float* accum = (float*)d_ws;
    float* out   = (float*)d_out;

    nq_init_kernel<<<1, 1, 0, stream>>>(accum);

    dim3 grid(WDIM / 16, HDIM / 16, BATCH);   // 12 x 12 x 4 = 576 blocks
    nq_main_kernel<<<grid, 256, 0, stream>>>(recon, target, accum);

    nq_final_kernel<<<1, 1, 0, stream>>>(accum, vq, out);
}